// Embedding_8555574853999
// MI455X (gfx1250) — compile-verified
//
#include <hip/hip_runtime.h>
#include <hip/hip_bf16.h>

// Problem constants from the reference
#define B_TOK   12800   // B*T = 64*200
#define KCODES  40
#define OTHERS  32
#define NCODE   526
#define DIM     512
#define MTILE   16      // tokens per workgroup
#define NTILES  (B_TOK / MTILE)   // 800

typedef __attribute__((ext_vector_type(2))) float v2f;
typedef __attribute__((ext_vector_type(4))) float v4f;
typedef __attribute__((ext_vector_type(8))) float v8f;

__global__ __launch_bounds__(256) void fused_multihot_gemm(
    const int*   __restrict__ code,    // [B_TOK, 40]
    const float* __restrict__ others,  // [B_TOK, 32]
    const float* __restrict__ W,       // [558, 512]
    const float* __restrict__ bias,    // [512]
    float*       __restrict__ out)     // [B_TOK, 512]
{
    const int tok0 = blockIdx.x * MTILE;
    const int tid  = threadIdx.x;

    __shared__ int   s_codes[MTILE * KCODES];          // 2.5 KB
    __shared__ float s_others[MTILE * OTHERS];         // 2.0 KB
    __shared__ __align__(16) float s_acc[MTILE * DIM]; // 32 KB

    // ---- Phase A: stage codes & dense features into LDS (coalesced) ----
    for (int i = tid; i < MTILE * KCODES; i += 256)
        s_codes[i] = code[tok0 * KCODES + i];
    for (int i = tid; i < MTILE * OTHERS; i += 256)
        s_others[i] = others[tok0 * OTHERS + i];
    __syncthreads();

    // ---- Phase B: dense part  [16 x 32] @ W[526:,:]  via V_WMMA_F32_16X16X4_F32 ----
    // wave w owns columns [w*64, w*64+64): 4 N-tiles of 16, K = 32 -> 8 wmma per tile.
    // A (16x4 f32): lanes 0-15 -> M=lane, VGPR{0,1}=K{k0,k0+1};
    //               lanes 16-31 -> M=lane-16, VGPR{0,1}=K{k0+2,k0+3}.
    // B (4x16 f32) mirrored: VGPR r, lane half h -> row k0+2h+r, col = lane%16.
    // C/D (16x16 f32): VGPR v, lane half h -> (M = v + 8h, N = lane%16).
    const int wave    = tid >> 5;
    const int lane    = tid & 31;
    const int laneHi  = lane >> 4;
    const int l16     = lane & 15;
    const int colbase = wave * 64;

    #pragma unroll
    for (int j = 0; j < 4; ++j) {
        const int n = colbase + j * 16 + l16;
        v8f c = {};
        #pragma unroll
        for (int k0 = 0; k0 < OTHERS; k0 += 4) {
            const int ka = k0 + 2 * laneHi;
            v2f a, bf;
            a.x  = s_others[l16 * OTHERS + ka];
            a.y  = s_others[l16 * OTHERS + ka + 1];
            bf.x = W[(NCODE + ka)     * DIM + n];
            bf.y = W[(NCODE + ka + 1) * DIM + n];
            c = __builtin_amdgcn_wmma_f32_16x16x4_f32(
                    /*neg_a=*/false, a, /*neg_b=*/false, bf,
                    /*c_mod=*/(short)0, c, /*reuse_a=*/false, /*reuse_b=*/false);
        }
        #pragma unroll
        for (int v = 0; v < 8; ++v)
            s_acc[(v + 8 * laneHi) * DIM + colbase + j * 16 + l16] = c[v];
    }
    __syncthreads();

    // ---- Phase C: gather-sum of 40 W rows per token + bias, natural layout ----
    // thread -> token tm = tid/16, column group g = tid%16; 8 passes of float4.
    const int tm = tid >> 4;
    const int g  = tid & 15;

    // Hoist the 40 row offsets into registers ONCE (full unroll -> SROA to VGPRs).
    int rowoff[KCODES];                       // offsets in v4f units (row * 128)
    #pragma unroll
    for (int k = 0; k < KCODES; ++k)
        rowoff[k] = s_codes[tm * KCODES + k] * (DIM / 4);

    const v4f* W4 = (const v4f*)W;
    const v4f* b4 = (const v4f*)bias;
    const v4f* s4 = (const v4f*)s_acc;
    v4f* out4 = (v4f*)out;

    for (int pass = 0; pass < 8; ++pass) {
        const int n4 = pass * 16 + g;                  // v4f column index (0..127)
        v4f acc = s4[tm * (DIM / 4) + n4] + b4[n4];    // wmma part + bias
        #pragma unroll
        for (int k = 0; k < KCODES; ++k)
            acc += W4[rowoff[k] + n4];                 // packed f32 adds (v_pk_add_f32)
        out4[(size_t)(tok0 + tm) * (DIM / 4) + n4] = acc;
    }
}

extern "C" void kernel_launch(void* const* d_in, const int* in_sizes, int n_in,
                              void* d_out, int out_size, void* d_ws, size_t ws_size,
                              hipStream_t stream) {
    const int*   code   = (const int*)d_in[0];
    const float* others = (const float*)d_in[1];
    // d_in[2] = length : unused (reference output ignores it)
    const float* W      = (const float*)d_in[3];
    const float* bias   = (const float*)d_in[4];
    float* out = (float*)d_out;

    fused_multihot_gemm<<<dim3(NTILES), dim3(256), 0, stream>>>(
        code, others, W, bias, out);
}